// DoyleSDE_26001732010511
// MI455X (gfx1250) — compile-verified
//
#include <hip/hip_runtime.h>

typedef __attribute__((ext_vector_type(2))) float v2f;
typedef __attribute__((ext_vector_type(4))) float v4f;
typedef __attribute__((ext_vector_type(8))) float v8f;

// ---- model constants (compile-time reciprocals: no runtime divides except 1/Rs) ----
#define K_C_AS 0.01016f
#define K_C_VS 0.65f
#define K_C_AP 0.0361f
#define K_C_VP 0.1408f
#define K_R_P  0.5f
#define K_R_S0 6.5f
#define K_A_RS 18.0f
#define K_V_TOT 5.058f
#define K_V_TO2 6.0f
#define K_O_2A 0.2f
#define K_M_0  0.36f
#define K_C_L  0.03f
#define K_C_R  0.05f

__device__ __forceinline__ v4f doyle_deriv(float Pas, float Pvs, float Pap, float O2v, float H)
{
    float Vvp = K_V_TOT - (K_C_AS * Pas + K_C_VS * Pvs + K_C_AP * Pap);
    float Pvp = Vvp * (1.0f / K_C_VP);
    float Rs  = fmaf(K_A_RS, O2v, K_R_S0);
    float rRs = __builtin_amdgcn_rcpf(Rs);          // one fast reciprocal per row
    float Fs  = (Pas - Pvs) * rRs;
    float Fp  = (Pap - Pvp) * (1.0f / K_R_P);
    float dO2 = K_O_2A - O2v;
    v4f r;
    r.x = (K_C_L * H * Pvp - Fs)        * (1.0f / K_C_AS);
    r.y = (Fs - K_C_R * H * Pvs)        * (1.0f / K_C_VS);
    r.z = (K_C_R * H * Pvs - Fp)        * (1.0f / K_C_AP);
    r.w = (Fs * dO2 - K_M_0)            * (1.0f / K_V_TO2);
    return r;
}

// One wave handles 32 rows per iteration:
//   2x V_WMMA_F32_16X16X4_F32 compute dot(x_row, K) for 16 rows each,
//   with A = K replicated (all 16 M-rows identical) and B = x-tile transposed,
//   so dot(row base+L) lands in lane L with no cross-lane traffic.
__global__ void __launch_bounds__(256)
doyle_wmma_kernel(const float* __restrict__ x,
                  const float* __restrict__ K,
                  const float* __restrict__ bias,
                  float* __restrict__ out,
                  int nRows, int rowStride /* = total lanes in grid */)
{
    const int lane   = threadIdx.x & 31;
    const int wave   = (blockIdx.x * blockDim.x + threadIdx.x) >> 5;
    const int laneM  = lane & 15;      // row within 16-row tile this lane feeds
    const int half   = lane >> 4;      // 0: supplies K=0,1 ; 1: supplies K=2,3

    const float k0 = K[0], k1 = K[1], k2 = K[2], k3 = K[3];
    const float b0 = bias[0];

    // A operand: K broadcast to 16x4. 32-bit A layout (16x4):
    //   lanes 0-15:  VGPR0 = A[M][0], VGPR1 = A[M][1]
    //   lanes 16-31: VGPR0 = A[M][2], VGPR1 = A[M][3]
    v2f amat;
    amat.x = (lane < 16) ? k0 : k2;
    amat.y = (lane < 16) ? k1 : k3;

    const int nFull = nRows & ~31;     // rows covered by full 32-row tiles

    // ---- main streaming loop (wave-uniform trip count -> EXEC all-ones at WMMA) ----
    for (long base = (long)wave * 32; base + 32 <= (long)nRows; base += rowStride) {
        // B operand: x-tile transposed (B[k][n] = x[row n][k]).
        //   lanes 0-15 supply K=0,1 of row (base+laneM); lanes 16-31 supply K=2,3.
        // Per-lane 8B loads tile 16 rows x 16B contiguously -> fully coalesced.
        const float* p0 = x + 4 * (base + laneM) + 2 * half;   // rows base..base+15
        v2f bmat0 = *(const v2f*)p0;
        v2f bmat1 = *(const v2f*)(p0 + 64);                    // rows base+16..base+31

        v8f cz = {};
        v8f d0 = __builtin_amdgcn_wmma_f32_16x16x4_f32(
                     false, amat, false, bmat0, (short)0, cz, false, false);
        v8f d1 = __builtin_amdgcn_wmma_f32_16x16x4_f32(
                     false, amat, false, bmat1, (short)0, cz, false, false);

        // All D rows identical (A rows identical) -> d[0] = dot(x[row N], K), N = lane%16.
        // Lane L (<16) owns row base+L via d0; lane L (>=16) owns row base+L via d1.
        float dot = (lane < 16) ? d0[0] : d1[0];
        float H   = b0 - dot;

        long row = base + lane;
        v4f xv = *(const v4f*)(x + 4 * row);    // hits WGP$ (lines just fetched above)
        v4f r  = doyle_deriv(xv.x, xv.y, xv.z, xv.w, H);
        __builtin_nontemporal_store(r, (v4f*)(out + 4 * row));  // streaming store
    }

    // ---- scalar tail (rows not forming a full 32-row tile; no WMMA here) ----
    int tid  = blockIdx.x * blockDim.x + threadIdx.x;
    int tail = nRows - nFull;
    if (tid < tail) {
        long row = (long)nFull + tid;
        v4f xv = *(const v4f*)(x + 4 * row);
        float dot = xv.x * k0 + xv.y * k1 + xv.z * k2 + xv.w * k3;
        v4f r = doyle_deriv(xv.x, xv.y, xv.z, xv.w, b0 - dot);
        __builtin_nontemporal_store(r, (v4f*)(out + 4 * row));
    }
}

extern "C" void kernel_launch(void* const* d_in, const int* in_sizes, int n_in,
                              void* d_out, int out_size, void* d_ws, size_t ws_size,
                              hipStream_t stream)
{
    (void)n_in; (void)d_ws; (void)ws_size; (void)out_size;
    // setup_inputs order: t, x, K, controlBias
    const float* x    = (const float*)d_in[1];
    const float* K    = (const float*)d_in[2];
    const float* bias = (const float*)d_in[3];
    float*       out  = (float*)d_out;

    const int nRows   = in_sizes[1] / 4;          // 16,777,216
    const int threads = 256;                      // 8 wave32 per block
    const int blocks  = 4096;                     // 32768 waves; stride 2^20 rows/pass
    const int stride  = blocks * threads;         // rows per grid pass (1 row/lane)

    doyle_wmma_kernel<<<blocks, threads, 0, stream>>>(x, K, bias, out, nRows, stride);
}